// ChessboardLayer_73280732004775
// MI455X (gfx1250) — compile-verified
//
#include <hip/hip_runtime.h>
#include <hip/hip_bf16.h>

// ChessboardLayer: space-to-depth (B=32,H=256,W=256,C=32) fp32.
// Pure permutation => 0 FLOPs, 512 MB of HBM traffic, ~22us floor @ 23.3 TB/s.
// Decomposes into 65536 contiguous 4KB block copies:
//   out chunk t (t = ((b*8+i)*8+j)*32 + r) of 1024 floats
//     == in[b, i*32+r, j*32:(j+1)*32, 0:32]  (contiguous 1024 floats).
// Strategy: CDNA5 async global->LDS->global DMA (ASYNCcnt path), 128-bit ops,
// 8 chunks in flight per block before each s_wait_asynccnt drain.
// Cache policy: data is touched exactly once and far exceeds the 192MB L2,
// so use TH=NT (cpol=1) on both the async loads and stores.

// Builtin parameter type (from hipcc diagnostic): int __vector_size__(16) *
// in addrspace(1) (global) and addrspace(3) (LDS).
typedef int v4i __attribute__((vector_size(16)));
typedef __attribute__((address_space(1))) v4i as1_v4i;
typedef __attribute__((address_space(3))) v4i as3_v4i;
typedef __attribute__((address_space(3))) void as3_void;

#define CPOL_TH_NT 1   // gfx12-family CPol: TH bits [2:0]; 1 = non-temporal

__device__ __forceinline__ void async_load_b128(const float4* gsrc, float4* lds_dst) {
#if __has_builtin(__builtin_amdgcn_global_load_async_to_lds_b128)
    __builtin_amdgcn_global_load_async_to_lds_b128(
        (as1_v4i*)gsrc, (as3_v4i*)lds_dst, 0, CPOL_TH_NT);
#else
    unsigned lds_off = (unsigned)(unsigned long long)(as3_void*)lds_dst;
    asm volatile("global_load_async_to_lds_b128 %0, %1, off th:TH_LOAD_NT"
                 :
                 : "v"(lds_off), "v"((unsigned long long)gsrc)
                 : "memory");
#endif
}

__device__ __forceinline__ void async_store_b128(float4* gdst, const float4* lds_src) {
#if __has_builtin(__builtin_amdgcn_global_store_async_from_lds_b128)
    __builtin_amdgcn_global_store_async_from_lds_b128(
        (as1_v4i*)gdst, (as3_v4i*)lds_src, 0, CPOL_TH_NT);
#else
    unsigned lds_off = (unsigned)(unsigned long long)(as3_void*)lds_src;
    asm volatile("global_store_async_from_lds_b128 %0, %1, off th:TH_STORE_NT"
                 :
                 : "v"((unsigned long long)gdst), "v"(lds_off)
                 : "memory");
#endif
}

__device__ __forceinline__ void wait_async0() {
#if __has_builtin(__builtin_amdgcn_s_wait_asynccnt)
    __builtin_amdgcn_s_wait_asynccnt(0);
#else
    asm volatile("s_wait_asynccnt 0" ::: "memory");
#endif
}

// Each block: 256 threads, 8 chunks of 4KB. Lane owns one float4 per chunk.
// Chunk id t = g*8 + k;  bits of t: r=t[4:0], j=t[7:5], i=t[10:8], b=t[15:11].
// Since k<8 and (g*8)%32 in {0,8,16,24}, all 8 chunks of a block share b,i,j.
__global__ __launch_bounds__(256) void ChessboardLayer_73280732004775_kernel(
    const float4* __restrict__ in, float4* __restrict__ out) {
    __shared__ float4 buf[8 * 256];   // 32 KB staging

    const unsigned tid = threadIdx.x;
    const unsigned g   = blockIdx.x;

    const unsigned j  = (g >> 2) & 7;
    const unsigned i  = (g >> 5) & 7;
    const unsigned b  =  g >> 8;
    const unsigned r0 = (g & 3) * 8;          // first r of this block's 8 chunks
    const unsigned h0 = i * 32 + r0;

    // float4 indices. Input row (b,h): base (b*256+h)*2048; col-band j: +j*256.
    const unsigned src4 = b * 524288u + h0 * 2048u + j * 256u + tid;
    // Output: chunk t starts at t*256 float4.
    const unsigned dst4 = g * 2048u + tid;

    float4* lslot = &buf[tid];

    // Phase 1: queue 8 async 128-bit loads (ASYNCcnt tracks them).
#pragma unroll
    for (int k = 0; k < 8; ++k)
        async_load_b128(in + (src4 + (unsigned)k * 2048u), lslot + k * 256);

    wait_async0();   // this wave's LDS slots are now filled

    // Phase 2: queue 8 async 128-bit stores from LDS to the packed output.
#pragma unroll
    for (int k = 0; k < 8; ++k)
        async_store_b128(out + (dst4 + (unsigned)k * 256u), lslot + k * 256);

    wait_async0();   // drain before wave teardown / LDS dealloc
}

extern "C" void kernel_launch(void* const* d_in, const int* in_sizes, int n_in,
                              void* d_out, int out_size, void* d_ws, size_t ws_size,
                              hipStream_t stream) {
    (void)in_sizes; (void)n_in; (void)d_ws; (void)ws_size; (void)out_size;
    const float4* in  = (const float4*)d_in[0];
    float4*       out = (float4*)d_out;
    // 65536 chunks / 8 per block = 8192 blocks of 256 threads (8 wave32s).
    ChessboardLayer_73280732004775_kernel<<<dim3(8192), dim3(256), 0, stream>>>(in, out);
}